// MeshDeformationBlock_56418690401097
// MI455X (gfx1250) — compile-verified
//
#include <hip/hip_runtime.h>
#include <hip/hip_bf16.h>

typedef __attribute__((ext_vector_type(2))) float v2f;
typedef __attribute__((ext_vector_type(8))) float v8f;

#define NEG_SLOPE 0.2f

// ---------------------------------------------------------------- utilities

__global__ void zero_kernel(float* __restrict__ p, long n) {
    long i = (long)blockIdx.x * blockDim.x + threadIdx.x;
    if (i < n) p[i] = 0.0f;
}

__global__ void init_softmax_kernel(float* __restrict__ emax, float* __restrict__ denom, int n) {
    int i = blockIdx.x * blockDim.x + threadIdx.x;
    if (i < n) {
        emax[i]  = -__builtin_inff();
        denom[i] = 0.0f;
    }
}

__device__ __forceinline__ void atomicMaxF(float* addr, float v) {
    // IEEE754 monotonic-bits trick: signed-int max for v>=0, unsigned min for v<0.
    if (v >= 0.0f) atomicMax((int*)addr, __float_as_int(v));
    else           atomicMin((unsigned int*)addr, __float_as_uint(v));
}

__device__ __forceinline__ v8f wmma_f32(v2f a, v2f b, v8f c) {
    return __builtin_amdgcn_wmma_f32_16x16x4_f32(false, a, false, b, (short)0, c, false, false);
}

// ---------------------------------------------------------------- GEMM (fp32 WMMA)
// Y[N,F] = X[N,K] @ W[K,F].  One wave computes a 16(M) x 64(N) strip: four 16x16
// accumulators share one A fragment per K-step (4x A reuse, 4 WMMAs in flight).
// f32 WMMA 16x16x4 operand layout (ISA 7.12.2):
//   A (16x4): lanes 0-15 hold M=lane, VGPR0=K0, VGPR1=K0+1; lanes 16-31 K0+2/K0+3
//   B (4x16): row striped across lanes within a VGPR, same K split as A
//   C/D (16x16 f32, 8 VGPRs): VGPR v -> row v (lanes 0-15), row v+8 (lanes 16-31)

template <int K, int F>
__global__ void gemm_wmma_f32(const float* __restrict__ X, const float* __restrict__ W,
                              float* __restrict__ Y, int N) {
    int wave = blockIdx.x * (blockDim.x >> 5) + (threadIdx.x >> 5);
    int lane = threadIdx.x & 31;
    constexpr int TN = 64;
    int tilesF = F / TN;
    int tiles  = (N >> 4) * tilesF;
    if (wave >= tiles) return;

    int m0 = (wave / tilesF) << 4;
    int n0 = (wave % tilesF) * TN;
    int half = lane >> 4;     // 0: K=k0,k0+1   1: K=k0+2,k0+3
    int l16  = lane & 15;

    const float* xrowBase = X + (size_t)(m0 + l16) * K;     // A: row m0+l16
    const float* xrow     = xrowBase + half * 2;            // pre-offset by K split
    const float* wbase    = W + n0 + l16;                   // B: col n0+l16 (+16j)

    v8f c0 = {}, c1 = {}, c2 = {}, c3 = {};

    constexpr int K4 = K & ~3;
    for (int k0 = 0; k0 < K4; k0 += 4) {
        v2f a;
        a.x = xrow[k0];
        a.y = xrow[k0 + 1];
        const float* wk = wbase + (size_t)(k0 + half * 2) * F;
        v2f b0, b1, b2, b3;
        b0.x = wk[0];      b0.y = wk[F];
        b1.x = wk[16];     b1.y = wk[F + 16];
        b2.x = wk[32];     b2.y = wk[F + 32];
        b3.x = wk[48];     b3.y = wk[F + 48];
        c0 = wmma_f32(a, b0, c0);
        c1 = wmma_f32(a, b1, c1);
        c2 = wmma_f32(a, b2, c2);
        c3 = wmma_f32(a, b3, c3);
    }

    if constexpr ((K & 3) != 0) {
        // One-time tail: clamp addresses (always valid), select-to-zero after the
        // load (v_cndmask, no EXEC divergence).
        int ka  = K4 + half * 2;
        int kc0 = (ka     < K) ? ka     : (K - 1);
        int kc1 = (ka + 1 < K) ? ka + 1 : (K - 1);
        bool v0 = (ka     < K);
        bool v1 = (ka + 1 < K);
        v2f a;
        a.x = v0 ? xrowBase[kc0] : 0.0f;
        a.y = v1 ? xrowBase[kc1] : 0.0f;
        const float* w0 = wbase + (size_t)kc0 * F;
        const float* w1 = wbase + (size_t)kc1 * F;
        v2f b0, b1, b2, b3;
        b0.x = v0 ? w0[0]  : 0.0f;  b0.y = v1 ? w1[0]  : 0.0f;
        b1.x = v0 ? w0[16] : 0.0f;  b1.y = v1 ? w1[16] : 0.0f;
        b2.x = v0 ? w0[32] : 0.0f;  b2.y = v1 ? w1[32] : 0.0f;
        b3.x = v0 ? w0[48] : 0.0f;  b3.y = v1 ? w1[48] : 0.0f;
        c0 = wmma_f32(a, b0, c0);
        c1 = wmma_f32(a, b1, c1);
        c2 = wmma_f32(a, b2, c2);
        c3 = wmma_f32(a, b3, c3);
    }

    float* y = Y + (size_t)m0 * F + n0 + l16;
#pragma unroll
    for (int v = 0; v < 8; ++v) {
        size_t row = (size_t)(v + half * 8) * F;
        y[row]      = c0[v];
        y[row + 16] = c1[v];
        y[row + 32] = c2[v];
        y[row + 48] = c3[v];
    }
}

// Scalar GEMM for the tiny last layer (F=3, not a multiple of 16).
__global__ void gemm_scalar(const float* __restrict__ X, const float* __restrict__ W,
                            float* __restrict__ Y, int N, int K, int F) {
    long i = (long)blockIdx.x * blockDim.x + threadIdx.x;
    if (i >= (long)N * F) return;
    int n = (int)(i / F), f = (int)(i % F);
    const float* x = X + (size_t)n * K;
    float s = 0.0f;
    for (int k = 0; k < K; ++k) s += x[k] * W[(size_t)k * F + f];
    Y[i] = s;
}

// ---------------------------------------------------------------- edge kernels
// Edge j < E comes from edge_index; j in [E, E+N) is the self-loop of node j-E.

__device__ __forceinline__ float lrelu(float h) {
    return (h > 0.0f) ? h : NEG_SLOPE * h;
}

// Vectorized score: requires F % 128 == 0 (float4 per lane, full wave coverage).
__global__ void edge_score_v4(const float* __restrict__ xl, const float* __restrict__ xr,
                              const float* __restrict__ att,
                              const int* __restrict__ src, const int* __restrict__ dst,
                              int E, int N, int F, float* __restrict__ e) {
    int wave = blockIdx.x * (blockDim.x >> 5) + (threadIdx.x >> 5);
    int lane = threadIdx.x & 31;
    int Etot = E + N;
    if (wave >= Etot) return;
    int s, d;
    if (wave < E) { s = src[wave]; d = dst[wave]; } else { s = d = wave - E; }
    const float4* pl = (const float4*)(xl + (size_t)s * F);
    const float4* pr = (const float4*)(xr + (size_t)d * F);
    const float4* pa = (const float4*)att;
    int n4 = F >> 2;
    float sum = 0.0f;
    for (int i = lane; i < n4; i += 32) {
        float4 a = pl[i];
        float4 b = pr[i];
        float4 w = pa[i];
        sum += lrelu(a.x + b.x) * w.x + lrelu(a.y + b.y) * w.y +
               lrelu(a.z + b.z) * w.z + lrelu(a.w + b.w) * w.w;
    }
#pragma unroll
    for (int off = 16; off > 0; off >>= 1) sum += __shfl_xor(sum, off, 32);
    if (lane == 0) e[wave] = sum;
}

// Scalar score for F=3.
__global__ void edge_score_small(const float* __restrict__ xl, const float* __restrict__ xr,
                                 const float* __restrict__ att,
                                 const int* __restrict__ src, const int* __restrict__ dst,
                                 int E, int N, int F, float* __restrict__ e) {
    int i = blockIdx.x * blockDim.x + threadIdx.x;
    int Etot = E + N;
    if (i >= Etot) return;
    int s, d;
    if (i < E) { s = src[i]; d = dst[i]; } else { s = d = i - E; }
    const float* pl = xl + (size_t)s * F;
    const float* pr = xr + (size_t)d * F;
    float sum = 0.0f;
    for (int f = 0; f < F; ++f) sum += lrelu(pl[f] + pr[f]) * att[f];
    e[i] = sum;
}

__global__ void edge_max_kernel(const float* __restrict__ e, const int* __restrict__ dst,
                                int E, int N, float* __restrict__ emax) {
    int i = blockIdx.x * blockDim.x + threadIdx.x;
    int Etot = E + N;
    if (i >= Etot) return;
    int d = (i < E) ? dst[i] : (i - E);
    atomicMaxF(&emax[d], e[i]);
}

__global__ void edge_exp_kernel(const float* __restrict__ e, const int* __restrict__ dst,
                                int E, int N, const float* __restrict__ emax,
                                float* __restrict__ eexp, float* __restrict__ denom) {
    int i = blockIdx.x * blockDim.x + threadIdx.x;
    int Etot = E + N;
    if (i >= Etot) return;
    int d = (i < E) ? dst[i] : (i - E);
    float v = expf(e[i] - emax[d]);
    eexp[i] = v;
    atomicAdd(&denom[d], v);
}

// Vectorized aggregate: requires F % 128 == 0.
__global__ void edge_aggregate_v4(const float* __restrict__ xl, const float* __restrict__ eexp,
                                  const float* __restrict__ denom,
                                  const int* __restrict__ src, const int* __restrict__ dst,
                                  int E, int N, int F, float* __restrict__ acc) {
    int wave = blockIdx.x * (blockDim.x >> 5) + (threadIdx.x >> 5);
    int lane = threadIdx.x & 31;
    int Etot = E + N;
    if (wave >= Etot) return;
    int s, d;
    if (wave < E) { s = src[wave]; d = dst[wave]; } else { s = d = wave - E; }
    float alpha = eexp[wave] / denom[d];
    const float4* pl = (const float4*)(xl + (size_t)s * F);
    float* pa = acc + (size_t)d * F;
    int n4 = F >> 2;
    for (int i = lane; i < n4; i += 32) {
        float4 v = pl[i];
        atomicAdd(&pa[4 * i + 0], alpha * v.x);
        atomicAdd(&pa[4 * i + 1], alpha * v.y);
        atomicAdd(&pa[4 * i + 2], alpha * v.z);
        atomicAdd(&pa[4 * i + 3], alpha * v.w);
    }
}

// Scalar aggregate for F=3.
__global__ void edge_aggregate_small(const float* __restrict__ xl, const float* __restrict__ eexp,
                                     const float* __restrict__ denom,
                                     const int* __restrict__ src, const int* __restrict__ dst,
                                     int E, int N, int F, float* __restrict__ acc) {
    int i = blockIdx.x * blockDim.x + threadIdx.x;
    int Etot = E + N;
    if (i >= Etot) return;
    int s, d;
    if (i < E) { s = src[i]; d = dst[i]; } else { s = d = i - E; }
    float alpha = eexp[i] / denom[d];
    const float* pl = xl + (size_t)s * F;
    float* pa = acc + (size_t)d * F;
    for (int f = 0; f < F; ++f) atomicAdd(&pa[f], alpha * pl[f]);
}

__global__ void finalize_kernel(const float* __restrict__ acc, const float* __restrict__ bias,
                                float* __restrict__ dest, long n, int F, int do_relu) {
    long i = (long)blockIdx.x * blockDim.x + threadIdx.x;
    if (i >= n) return;
    int f = (int)(i % F);
    float v = acc[i] + bias[f];
    if (do_relu) v = fmaxf(v, 0.0f);
    dest[i] = v;
}

// ---------------------------------------------------------------- host driver

static inline int cdiv(long a, long b) { return (int)((a + b - 1) / b); }

struct EdgeBufs {
    float *e, *eexp, *emax, *denom;
    const int *src, *dst;
    int E, N;
};

static void edge_phase(const float* xl, const float* xr, const float* att, const float* b,
                       float* acc, float* dest, int Fout, int do_relu,
                       const EdgeBufs& eb, hipStream_t stream) {
    int N = eb.N, E = eb.E;
    long nf = (long)N * Fout;
    int Etot = E + N;
    int waveBlocks = cdiv((long)Etot * 32, 256);
    int thrBlocks  = cdiv(Etot, 256);

    zero_kernel<<<cdiv(nf, 256), 256, 0, stream>>>(acc, nf);
    init_softmax_kernel<<<cdiv(N, 256), 256, 0, stream>>>(eb.emax, eb.denom, N);

    if ((Fout & 127) == 0) {
        edge_score_v4<<<waveBlocks, 256, 0, stream>>>(xl, xr, att, eb.src, eb.dst, E, N, Fout, eb.e);
    } else {
        edge_score_small<<<thrBlocks, 256, 0, stream>>>(xl, xr, att, eb.src, eb.dst, E, N, Fout, eb.e);
    }
    edge_max_kernel<<<thrBlocks, 256, 0, stream>>>(eb.e, eb.dst, E, N, eb.emax);
    edge_exp_kernel<<<thrBlocks, 256, 0, stream>>>(eb.e, eb.dst, E, N, eb.emax, eb.eexp, eb.denom);
    if ((Fout & 127) == 0) {
        edge_aggregate_v4<<<waveBlocks, 256, 0, stream>>>(xl, eb.eexp, eb.denom, eb.src, eb.dst, E, N, Fout, acc);
    } else {
        edge_aggregate_small<<<thrBlocks, 256, 0, stream>>>(xl, eb.eexp, eb.denom, eb.src, eb.dst, E, N, Fout, acc);
    }
    finalize_kernel<<<cdiv(nf, 256), 256, 0, stream>>>(acc, b, dest, nf, Fout, do_relu);
}

template <int K, int F>
static void launch_gemm_pair(const float* act, const float* Wl, const float* Wr,
                             float* xl, float* xr, int N, hipStream_t stream) {
    long waves = ((long)N / 16) * (F / 64);
    int blocks = cdiv(waves * 32, 256);
    gemm_wmma_f32<K, F><<<blocks, 256, 0, stream>>>(act, Wl, xl, N);
    gemm_wmma_f32<K, F><<<blocks, 256, 0, stream>>>(act, Wr, xr, N);
}

extern "C" void kernel_launch(void* const* d_in, const int* in_sizes, int n_in,
                              void* d_out, int out_size, void* d_ws, size_t ws_size,
                              hipStream_t stream) {
    const float* x          = (const float*)d_in[0];
    const int*   edge_index = (const int*)d_in[1];
    int N = in_sizes[0] / 39;
    int E = in_sizes[1] / 2;

    const float* Wl1  = (const float*)d_in[2];
    const float* Wr1  = (const float*)d_in[3];
    const float* att1 = (const float*)d_in[4];
    const float* b1   = (const float*)d_in[5];
    const float* Wl21 = (const float*)d_in[6];
    const float* Wr21 = (const float*)d_in[7];
    const float* att21= (const float*)d_in[8];
    const float* b21  = (const float*)d_in[9];
    const float* Wl22 = (const float*)d_in[10];
    const float* Wr22 = (const float*)d_in[11];
    const float* att22= (const float*)d_in[12];
    const float* b22  = (const float*)d_in[13];
    const float* Wl23 = (const float*)d_in[14];
    const float* Wr23 = (const float*)d_in[15];
    const float* att23= (const float*)d_in[16];
    const float* b23  = (const float*)d_in[17];
    const float* Wl3  = (const float*)d_in[18];
    const float* Wr3  = (const float*)d_in[19];
    const float* att3 = (const float*)d_in[20];
    const float* b3   = (const float*)d_in[21];

    float* ws = (float*)d_ws;
    size_t NB = (size_t)N * 1024;
    int Etot = E + N;
    float* bufA  = ws;
    float* bufB  = bufA + NB;
    float* xl    = bufB + NB;
    float* xr    = xl + NB;
    float* e     = xr + NB;
    float* eexp  = e + Etot;
    float* emax  = eexp + Etot;
    float* denom = emax + N;

    float* out    = (float*)d_out;            // [N,128]
    float* coords = out + (size_t)N * 128;    // [N,3]

    EdgeBufs eb{e, eexp, emax, denom, edge_index, edge_index + E, E, N};

    // Layer 1: 39 -> 1024
    launch_gemm_pair<39, 1024>(x, Wl1, Wr1, xl, xr, N, stream);
    edge_phase(xl, xr, att1, b1, bufA, bufA, 1024, 1, eb, stream);
    // Layer 2: 1024 -> 512
    launch_gemm_pair<1024, 512>(bufA, Wl21, Wr21, xl, xr, N, stream);
    edge_phase(xl, xr, att21, b21, bufB, bufB, 512, 1, eb, stream);
    // Layer 3: 512 -> 256
    launch_gemm_pair<512, 256>(bufB, Wl22, Wr22, xl, xr, N, stream);
    edge_phase(xl, xr, att22, b22, bufA, bufA, 256, 1, eb, stream);
    // Layer 4: 256 -> 128  (result is first output, also input to layer 5)
    launch_gemm_pair<256, 128>(bufA, Wl23, Wr23, xl, xr, N, stream);
    edge_phase(xl, xr, att23, b23, bufB, out, 128, 1, eb, stream);
    // Layer 5: 128 -> 3 (coords, no relu, scalar GEMM since Fout=3)
    {
        int blocks = cdiv((long)N * 3, 256);
        gemm_scalar<<<blocks, 256, 0, stream>>>(out, Wl3, xl, N, 128, 3);
        gemm_scalar<<<blocks, 256, 0, stream>>>(out, Wr3, xr, N, 128, 3);
        edge_phase(xl, xr, att3, b3, bufA, coords, 3, 0, eb, stream);
    }
}